// DynamicsNetwork_36988258353597
// MI455X (gfx1250) — compile-verified
//
#include <hip/hip_runtime.h>
#include <hip/hip_bf16.h>

// ---------------------------------------------------------------------------
// Problem constants (match reference)
// D=2, C=1, PDE=4, L1=16, L2=32, L3=48, LATENT=16; phase-1 input width = 17,
// phase-3 input width = 15.
// ---------------------------------------------------------------------------

typedef __attribute__((ext_vector_type(2))) float v2f;
typedef __attribute__((ext_vector_type(4))) float v4f;
typedef __attribute__((ext_vector_type(8))) float v8f;

__device__ __forceinline__ v8f wmma4(v2f a, v2f b, v8f c) {
  // D(16x16,f32) = A(16x4,f32) * B(4x16,f32) + C
  return __builtin_amdgcn_wmma_f32_16x16x4_f32(false, a, false, b, (short)0, c,
                                               false, false);
}

// Branch-free tanh: single-instruction v_tanh_f32 when available, else
// tanh(x) = 1 - 2/(exp2(2*log2e*x)+1)  (exact at +-inf, NaN-propagating).
__device__ __forceinline__ float fast_tanh(float x) {
#if __has_builtin(__builtin_amdgcn_tanhf)
  return __builtin_amdgcn_tanhf(x);
#else
  float t = x * 2.8853900817779268f;  // 2*log2(e)
#if __has_builtin(__builtin_amdgcn_exp2f)
  float e = __builtin_amdgcn_exp2f(t);
#else
  float e = exp2f(t);
#endif
#if __has_builtin(__builtin_amdgcn_rcpf)
  float r = __builtin_amdgcn_rcpf(e + 1.0f);
#else
  float r = 1.0f / (e + 1.0f);
#endif
  return __builtin_fmaf(-2.0f, r, 1.0f);
#endif
}

// ---------------------------------------------------------------------------
// Pre-swizzle weight fragments into LDS.
// Fragment (kb, j): lane<16  -> {B[kb*4+0][j*16+n], B[kb*4+1][j*16+n]}
//                   lane>=16 -> {B[kb*4+2][j*16+n], B[kb*4+3][j*16+n]}
// where B[k][n] = W[n*Kin + k]  (out = h @ W.T), zero-padded past Kin.
// ---------------------------------------------------------------------------
__device__ inline void fill_frags(float* frag, const float* W, int Kin, int KCH,
                                  int NBLK, int tid, int nth) {
  int total = KCH * NBLK * 32;
  for (int e = tid; e < total; e += nth) {
    int lane = e & 31;
    int idx  = e >> 5;
    int kb   = idx / NBLK;
    int j    = idx - kb * NBLK;
    int half = lane >> 4, n = lane & 15;
    int k0   = kb * 4 + half * 2;
    int col  = j * 16 + n;
    float v0 = (k0     < Kin) ? W[col * Kin + k0]     : 0.f;
    float v1 = (k0 + 1 < Kin) ? W[col * Kin + k0 + 1] : 0.f;
    frag[(idx * 32 + lane) * 2 + 0] = v0;
    frag[(idx * 32 + lane) * 2 + 1] = v1;
  }
}

// One dense layer over a 16-row LDS tile: acc[j] = Atile x Bfrags (16 cols each)
template <int KCH, int NBLK>
__device__ __forceinline__ void mlp_matmul(const float* hbuf, int rs,
                                           const float* frag, int lane,
                                           v8f acc[NBLK]) {
  const int half = lane >> 4, n = lane & 15;
  v2f A[KCH];
#pragma unroll
  for (int kb = 0; kb < KCH; ++kb)
    A[kb] = *(const v2f*)(hbuf + n * rs + kb * 4 + half * 2);
  const v8f zero = {0.f, 0.f, 0.f, 0.f, 0.f, 0.f, 0.f, 0.f};
#pragma unroll
  for (int j = 0; j < NBLK; ++j) {
    v8f a = zero;
#pragma unroll
    for (int kb = 0; kb < KCH; ++kb) {
      v2f b = *(const v2f*)(frag + ((kb * NBLK + j) * 32 + lane) * 2);
      a = wmma4(A[kb], b, a);
    }
    acc[j] = a;
  }
}

// bias (+ optional tanh) and write back into the LDS tile (C/D layout:
// col = lane&15, row = r + 8*(lane>>4))
template <int NBLK, bool TANH>
__device__ __forceinline__ void mlp_store(const v8f acc[NBLK], const float* bias,
                                          float* hout, int rs, int lane) {
  const int half = lane >> 4, n = lane & 15;
#pragma unroll
  for (int j = 0; j < NBLK; ++j) {
    float bv = bias[j * 16 + n];
#pragma unroll
    for (int r = 0; r < 8; ++r) {
      float v = acc[j][r] + bv;
      if (TANH) v = fast_tanh(v);
      hout[(r + half * 8) * rs + j * 16 + n] = v;
    }
  }
}

// ---------------------------------------------------------------------------
// Kernel 0: zero the workspace accumulator / T region (ws is poisoned 0xAA)
// ---------------------------------------------------------------------------
__global__ void k0_init(float* ws) {
  if (threadIdx.x < 45) ws[threadIdx.x] = 0.f;
}

// ---------------------------------------------------------------------------
// Kernel 1: phase-1 MLP (17->16->32->16, tanh) + column-sum reduction to ws[0:16)
// ---------------------------------------------------------------------------
#define NWAVES 8
#define RS1 36  // row stride (floats); max cols = 32, padded input = 20

__global__ void k1_reduce(const float* __restrict__ means,
                          const float* __restrict__ cov,
                          const float* __restrict__ u,
                          const float* __restrict__ bnd,
                          const float* __restrict__ su,
                          const float* __restrict__ sux,
                          const float* __restrict__ suxx,
                          const float* __restrict__ spde,
                          const float* __restrict__ lw1, const float* __restrict__ lb1,
                          const float* __restrict__ lw2, const float* __restrict__ lb2,
                          const float* __restrict__ lw3, const float* __restrict__ lb3,
                          float* __restrict__ latentSum, int ntiles) {
  __shared__ float frag[21 * 64];          // 5 + 8 + 8 fragments
  __shared__ float hbuf[NWAVES][16 * RS1];
  __shared__ float blockAcc[16];

  const int tid = threadIdx.x, nth = blockDim.x;
  fill_frags(frag,            lw1, 17, 5, 1, tid, nth);  // 17->16  (K pad 20)
  fill_frags(frag + 5 * 64,   lw2, 16, 4, 2, tid, nth);  // 16->32
  fill_frags(frag + 13 * 64,  lw3, 32, 8, 1, tid, nth);  // 32->16
  if (tid < 16) blockAcc[tid] = 0.f;
  __syncthreads();

  const int wave = tid >> 5, lane = tid & 31;
  float* hb = hbuf[wave];
  const int gw = blockIdx.x * NWAVES + wave;
  const int nw = gridDim.x * NWAVES;

  for (int tile = gw; tile < ntiles; tile += nw) {
    if (lane < 16) {  // each of 16 lanes fills one row of the input tile
      size_t row = (size_t)tile * 16 + lane;
      float* p = hb + lane * RS1;
      v2f mv  = *(const v2f*)(means + row * 2);
      v4f cv  = *(const v4f*)(cov + row * 4);
      v2f xv  = *(const v2f*)(sux + row * 2);
      v2f xxv = *(const v2f*)(suxx + row * 2);
      v4f pv  = *(const v4f*)(spde + row * 4);
      p[0]  = mv[0];  p[1]  = mv[1];
      p[2]  = cv[0];  p[3]  = cv[1];  p[4] = cv[2];  p[5] = cv[3];
      p[6]  = u[row];
      p[7]  = bnd[row];
      p[8]  = su[row];
      p[9]  = xv[0];  p[10] = xv[1];
      p[11] = xxv[0]; p[12] = xxv[1];
      p[13] = pv[0];  p[14] = pv[1];  p[15] = pv[2]; p[16] = pv[3];
      p[17] = 0.f; p[18] = 0.f; p[19] = 0.f;  // K padding
    }
    v8f a1[1]; mlp_matmul<5, 1>(hb, RS1, frag,           lane, a1);
    mlp_store<1, true>(a1, lb1, hb, RS1, lane);
    v8f a2[2]; mlp_matmul<4, 2>(hb, RS1, frag + 5 * 64,  lane, a2);
    mlp_store<2, true>(a2, lb2, hb, RS1, lane);
    v8f a3[1]; mlp_matmul<8, 1>(hb, RS1, frag + 13 * 64, lane, a3);
    // tanh + partial column sum (lane and lane+16 cover all 16 rows of col n)
    float bv = lb3[lane & 15];
    float s = 0.f;
#pragma unroll
    for (int r = 0; r < 8; ++r) s += fast_tanh(a3[0][r] + bv);
    atomicAdd(&blockAcc[lane & 15], s);
  }
  __syncthreads();
  if (tid < 16) atomicAdd(&latentSum[tid], blockAcc[tid]);
}

// ---------------------------------------------------------------------------
// Kernel 2: five tiny T-nets, one per block: latent(16)->48->32->dd^2 (+I)
// ---------------------------------------------------------------------------
struct K2Args {
  const float* w1[5]; const float* b1[5];
  const float* w2[5]; const float* b2[5];
  const float* w3[5]; const float* b3[5];
  int dd[5]; int off[5];
  const float* latentSum; float* ws;
  float invN;
};

__global__ void k2_tnets(K2Args a) {
  const int t = blockIdx.x, tid = threadIdx.x;
  __shared__ float lat[16], h1[48], h2[32];
  if (tid < 16) lat[tid] = a.latentSum[tid] * a.invN;
  __syncthreads();
  if (tid < 48) {
    float s = a.b1[t][tid];
    const float* w = a.w1[t] + tid * 16;
    for (int k = 0; k < 16; ++k) s += w[k] * lat[k];
    h1[tid] = fast_tanh(s);
  }
  __syncthreads();
  if (tid < 32) {
    float s = a.b2[t][tid];
    const float* w = a.w2[t] + tid * 48;
    for (int k = 0; k < 48; ++k) s += w[k] * h1[k];
    h2[tid] = fast_tanh(s);
  }
  __syncthreads();
  const int dd = a.dd[t], nn = dd * dd;
  if (tid < nn) {
    float s = a.b3[t][tid];
    const float* w = a.w3[t] + tid * 32;
    for (int k = 0; k < 32; ++k) s += w[k] * h2[k];
    if ((tid / dd) == (tid % dd)) s += 1.f;  // + identity
    a.ws[16 + a.off[t] + tid] = s;
  }
}

// ---------------------------------------------------------------------------
// Kernel 3: apply T transforms, MLP 15->16->32->48->16, write (N,16)
// ---------------------------------------------------------------------------
#define RS3 52  // max cols = 48

__global__ void k3_out(const float* __restrict__ cov,
                       const float* __restrict__ u,
                       const float* __restrict__ bnd,
                       const float* __restrict__ su,
                       const float* __restrict__ sux,
                       const float* __restrict__ suxx,
                       const float* __restrict__ spde,
                       const float* __restrict__ Tm,  // ws+16: 29 floats
                       const float* __restrict__ jw1, const float* __restrict__ jb1,
                       const float* __restrict__ jw2, const float* __restrict__ jb2,
                       const float* __restrict__ jw3, const float* __restrict__ jb3,
                       const float* __restrict__ jw4, const float* __restrict__ jb4,
                       float* __restrict__ out, int ntiles) {
  __shared__ float frag[48 * 64];  // 4 + 8 + 24 + 12 fragments
  __shared__ float hbuf[NWAVES][16 * RS3];
  __shared__ float Ts[29];

  const int tid = threadIdx.x, nth = blockDim.x;
  fill_frags(frag,            jw1, 15, 4,  1, tid, nth);  // 15->16 (K pad 16)
  fill_frags(frag + 4 * 64,   jw2, 16, 4,  2, tid, nth);  // 16->32
  fill_frags(frag + 12 * 64,  jw3, 32, 8,  3, tid, nth);  // 32->48
  fill_frags(frag + 36 * 64,  jw4, 48, 12, 1, tid, nth);  // 48->16
  if (tid < 29) Ts[tid] = Tm[tid];
  __syncthreads();

  const int wave = tid >> 5, lane = tid & 31;
  float* hb = hbuf[wave];
  const int gw = blockIdx.x * NWAVES + wave;
  const int nw = gridDim.x * NWAVES;

  for (int tile = gw; tile < ntiles; tile += nw) {
    if (lane < 16) {
      size_t row = (size_t)tile * 16 + lane;
      float* p = hb + lane * RS3;
      // t_cov = T @ covmat
      float T00 = Ts[0], T01 = Ts[1], T10 = Ts[2], T11 = Ts[3];
      v4f cv = *(const v4f*)(cov + row * 4);
      p[0] = T00 * cv[0] + T01 * cv[2];
      p[1] = T00 * cv[1] + T01 * cv[3];
      p[2] = T10 * cv[0] + T11 * cv[2];
      p[3] = T10 * cv[1] + T11 * cv[3];
      float Tu = Ts[4];
      p[4] = u[row] * Tu;
      p[5] = bnd[row];
      p[6] = su[row] * Tu;
      v2f xv = *(const v2f*)(sux + row * 2);
      p[7] = Ts[5] * xv[0] + Ts[6] * xv[1];
      p[8] = Ts[7] * xv[0] + Ts[8] * xv[1];
      v2f xxv = *(const v2f*)(suxx + row * 2);
      p[9]  = Ts[9]  * xxv[0] + Ts[10] * xxv[1];
      p[10] = Ts[11] * xxv[0] + Ts[12] * xxv[1];
      v4f pv = *(const v4f*)(spde + row * 4);
#pragma unroll
      for (int i = 0; i < 4; ++i)
        p[11 + i] = Ts[13 + i * 4 + 0] * pv[0] + Ts[13 + i * 4 + 1] * pv[1] +
                    Ts[13 + i * 4 + 2] * pv[2] + Ts[13 + i * 4 + 3] * pv[3];
      p[15] = 0.f;  // K padding
    }
    v8f a1[1]; mlp_matmul<4, 1>(hb, RS3, frag,           lane, a1);
    mlp_store<1, true>(a1, jb1, hb, RS3, lane);
    v8f a2[2]; mlp_matmul<4, 2>(hb, RS3, frag + 4 * 64,  lane, a2);
    mlp_store<2, true>(a2, jb2, hb, RS3, lane);
    v8f a3[3]; mlp_matmul<8, 3>(hb, RS3, frag + 12 * 64, lane, a3);
    mlp_store<3, true>(a3, jb3, hb, RS3, lane);
    v8f a4[1]; mlp_matmul<12, 1>(hb, RS3, frag + 36 * 64, lane, a4);
    // final layer: bias only, store straight to global (coalesced 64B runs)
    const int half = lane >> 4, n = lane & 15;
    float bv = jb4[n];
#pragma unroll
    for (int r = 0; r < 8; ++r)
      out[((size_t)tile * 16 + r + half * 8) * 16 + n] = a4[0][r] + bv;
  }
}

// ---------------------------------------------------------------------------
// Host launcher
// ---------------------------------------------------------------------------
extern "C" void kernel_launch(void* const* d_in, const int* in_sizes, int n_in,
                              void* d_out, int out_size, void* d_ws, size_t ws_size,
                              hipStream_t stream) {
  const float* means = (const float*)d_in[0];
  const float* cov   = (const float*)d_in[1];
  const float* u     = (const float*)d_in[2];
  const float* bnd   = (const float*)d_in[3];
  const float* su    = (const float*)d_in[4];
  const float* sux   = (const float*)d_in[5];
  const float* suxx  = (const float*)d_in[6];
  const float* spde  = (const float*)d_in[7];
  const float* lw1 = (const float*)d_in[8],  *lb1 = (const float*)d_in[9];
  const float* lw2 = (const float*)d_in[10], *lb2 = (const float*)d_in[11];
  const float* lw3 = (const float*)d_in[12], *lb3 = (const float*)d_in[13];
  // t, u, x, xx, p nets: 6 tensors each starting at index 14
  const float* jw1 = (const float*)d_in[44], *jb1 = (const float*)d_in[45];
  const float* jw2 = (const float*)d_in[46], *jb2 = (const float*)d_in[47];
  const float* jw3 = (const float*)d_in[48], *jb3 = (const float*)d_in[49];
  const float* jw4 = (const float*)d_in[50], *jb4 = (const float*)d_in[51];

  float* ws  = (float*)d_ws;
  float* out = (float*)d_out;

  const int N = in_sizes[3];      // boundaries has N elements
  const int ntiles = N / 16;      // N = 1,000,000 -> 62500 exactly

  k0_init<<<1, 64, 0, stream>>>(ws);

  int grid1 = (ntiles + NWAVES - 1) / NWAVES;
  if (grid1 > 2048) grid1 = 2048;
  k1_reduce<<<grid1, 32 * NWAVES, 0, stream>>>(
      means, cov, u, bnd, su, sux, suxx, spde,
      lw1, lb1, lw2, lb2, lw3, lb3, ws, ntiles);

  K2Args a;
  static const int dds[5]  = {2, 1, 2, 2, 4};
  static const int offs[5] = {0, 4, 5, 9, 13};
  for (int t = 0; t < 5; ++t) {
    int base = 14 + t * 6;
    a.w1[t] = (const float*)d_in[base + 0];
    a.b1[t] = (const float*)d_in[base + 1];
    a.w2[t] = (const float*)d_in[base + 2];
    a.b2[t] = (const float*)d_in[base + 3];
    a.w3[t] = (const float*)d_in[base + 4];
    a.b3[t] = (const float*)d_in[base + 5];
    a.dd[t] = dds[t];
    a.off[t] = offs[t];
  }
  a.latentSum = ws;
  a.ws = ws;
  a.invN = 1.0f / (float)N;
  k2_tnets<<<5, 64, 0, stream>>>(a);

  int grid3 = (ntiles + NWAVES - 1) / NWAVES;
  if (grid3 > 2048) grid3 = 2048;
  k3_out<<<grid3, 32 * NWAVES, 0, stream>>>(
      cov, u, bnd, su, sux, suxx, spde, ws + 16,
      jw1, jb1, jw2, jb2, jw3, jb3, jw4, jb4, out, ntiles);
}